// TokenEncoder_40922448396579
// MI455X (gfx1250) — compile-verified
//
#include <hip/hip_runtime.h>
#include <math.h>

typedef __attribute__((ext_vector_type(2))) float v2f;
typedef __attribute__((ext_vector_type(8))) float v8f;

#define NB    64      // B
#define NC    512     // C
#define NF    48      // F
#define DM    16      // D (model dim)
#define TN    49      // real tokens
#define TP    64      // padded tokens
#define NHEAD 4
#define DHD   4
#define NFF   64
#define NL    2
#define NHID  256
#define EPSL  1e-5f

// LDS strides (padded to dodge 64-bank conflicts on column walks)
#define XS 18   // x / h / ctx buffers (64 x 16 logical)
#define QS 50   // qkv buffer        (64 x 48 logical)
#define SS 66   // scores / ff buffer (64 x 64 logical)

// ---------- f32 16x16x4 WMMA fragment helpers (ISA 7.12.2 layouts) ----------
// A (16x4, row-major, stride in floats): VGPR0 = K{0|2}, VGPR1 = K{1|3}
__device__ __forceinline__ v2f frag_a4(const float* src, int stride, int lane) {
    int row = lane & 15;
    int kb  = (lane >> 4) << 1;
    v2f a;
    a.x = src[row * stride + kb];
    a.y = src[row * stride + kb + 1];
    return a;
}
// B (4x16, row-major K x N): VGPR0 = row K{0|2}, VGPR1 = row K{1|3}
__device__ __forceinline__ v2f frag_b4(const float* src, int stride, int lane) {
    int col = lane & 15;
    int kb  = (lane >> 4) << 1;
    v2f b;
    b.x = src[kb * stride + col];
    b.y = src[(kb + 1) * stride + col];
    return b;
}
// D/C (16x16): VGPR i = row i (lanes 0-15) / row i+8 (lanes 16-31)
__device__ __forceinline__ void frag_store(float* dst, int stride, int lane, v8f d) {
    int col   = lane & 15;
    int rbase = (lane >> 4) << 3;
#pragma unroll
    for (int i = 0; i < 8; ++i) dst[(rbase + i) * stride + col] = d[i];
}

#define WMMA4(a, b, c) \
    __builtin_amdgcn_wmma_f32_16x16x4_f32(false, (a), false, (b), (short)0, (c), false, false)

// row-local LayerNorm over D=16; each lane owns 2 rows of the 64-row tile
__device__ __forceinline__ void layernorm_lds(const float* src, float* dst,
                                              const float* s, const float* b, int lane) {
#pragma unroll
    for (int rr = 0; rr < 2; ++rr) {
        int row = (lane << 1) + rr;
        float sum = 0.f, sq = 0.f;
#pragma unroll
        for (int d = 0; d < DM; ++d) { float v = src[row * XS + d]; sum += v; sq += v * v; }
        float m   = sum * (1.0f / DM);
        float var = sq * (1.0f / DM) - m * m;
        float rs  = rsqrtf(var + EPSL);
#pragma unroll
        for (int d = 0; d < DM; ++d)
            dst[row * XS + d] = (src[row * XS + d] - m) * rs * s[d] + b[d];
    }
}

__device__ __forceinline__ float gelu_exact(float x) {
    return 0.5f * x * (1.0f + erff(x * 0.70710678118654752f));
}

// ============ Kernel 1: fused embed + 2-layer transformer, one wave/sequence ============
__global__ void __launch_bounds__(32)
seq_encoder_kernel(const float* __restrict__ couple,      // (B,F,C)
                   const float* __restrict__ feat_emb,    // (F,D)
                   const float* __restrict__ feat_bias,   // (F,D)
                   const float* __restrict__ cls_tok,     // (D)
                   const float* __restrict__ track_emb,   // (3,D)
                   const int*   __restrict__ role_ids,    // (F)
                   const float* __restrict__ ln1_s, const float* __restrict__ ln1_b,
                   const float* __restrict__ qkv_w, const float* __restrict__ qkv_b,
                   const float* __restrict__ out_w, const float* __restrict__ out_b,
                   const float* __restrict__ ln2_s, const float* __restrict__ ln2_b,
                   const float* __restrict__ ff1_w, const float* __restrict__ ff1_b,
                   const float* __restrict__ ff2_w, const float* __restrict__ ff2_b,
                   float* __restrict__ cls_out)            // (N,16) staging
{
    __shared__ float xb[TP * XS];    // residual stream
    __shared__ float hb[TP * XS];    // LN output, reused as ctx buffer
    __shared__ float qb[TP * QS];    // q|k|v (64 x 48)
    __shared__ float sb[TP * SS];    // attn scores / ff1 activations (64 x 64)

    const int lane = threadIdx.x;
    const int n    = blockIdx.x;          // sequence id, n = b*C + c
    const int b    = n >> 9;
    const int c    = n & (NC - 1);

    // ---- build x = [cls ; scalar*emb + bias + track] , zero-pad rows 49..63 ----
#pragma unroll
    for (int i = 0; i < 32; ++i) {
        int idx = i * 32 + lane;          // 0..1023 over 64x16
        int t = idx >> 4, d = idx & 15;
        float v;
        if (t == 0) {
            v = cls_tok[d];
        } else if (t <= NF) {
            int f = t - 1;
            float sc = couple[(b * NF + f) * NC + c];
            v = sc * feat_emb[f * DM + d] + feat_bias[f * DM + d]
                + track_emb[role_ids[f] * DM + d];
        } else {
            v = 0.0f;
        }
        xb[t * XS + d] = v;
    }
    __syncthreads();

    const float attn_scale = 0.5f;   // 1/sqrt(DH=4)

    for (int l = 0; l < NL; ++l) {
        // ---------------- LN1 ----------------
        layernorm_lds(xb, hb, ln1_s + l * DM, ln1_b + l * DM, lane);
        __syncthreads();

        // ---------------- qkv = h @ Wqkv + b  (64x16 @ 16x48) ----------------
        {
            const float* wl = qkv_w + l * DM * 3 * DM;   // (16,48)
            const float* bl = qkv_b + l * 3 * DM;
            int col = lane & 15;
#pragma unroll
            for (int mt = 0; mt < 4; ++mt)
#pragma unroll
                for (int nt = 0; nt < 3; ++nt) {
                    v8f acc;
                    float bv = bl[nt * 16 + col];
#pragma unroll
                    for (int i = 0; i < 8; ++i) acc[i] = bv;
#pragma unroll
                    for (int kk = 0; kk < 4; ++kk) {
                        v2f a = frag_a4(&hb[(mt * 16) * XS + kk * 4], XS, lane);
                        v2f bf = frag_b4(wl + (kk * 4) * 48 + nt * 16, 48, lane);
                        acc = WMMA4(a, bf, acc);
                    }
                    frag_store(&qb[(mt * 16) * QS + nt * 16], QS, lane, acc);
                }
        }
        __syncthreads();

        // ---------------- attention, head by head ----------------
        for (int hd = 0; hd < NHEAD; ++hd) {
            // scores: Q_h(64x4) @ K_h^T(4x64), single K-step (K = DH = 4)
#pragma unroll
            for (int qt = 0; qt < 4; ++qt)
#pragma unroll
                for (int kt = 0; kt < 4; ++kt) {
                    v2f a = frag_a4(&qb[(qt * 16) * QS + hd * 4], QS, lane);
                    // B[k'][ncol] = K[kt*16+ncol][hd*4+k']
                    v2f bf;
                    {
                        int col = lane & 15, kb = (lane >> 4) << 1;
                        const float* kp = &qb[(kt * 16 + col) * QS + DM + hd * 4];
                        bf.x = kp[kb]; bf.y = kp[kb + 1];
                    }
                    v8f acc = {};
                    acc = WMMA4(a, bf, acc);
                    // scaled store
                    int col = lane & 15, rb = (lane >> 4) << 3;
#pragma unroll
                    for (int i = 0; i < 8; ++i)
                        sb[(qt * 16 + rb + i) * SS + kt * 16 + col] = acc[i] * attn_scale;
                }
            __syncthreads();

            // softmax over keys; 2 rows per lane.
            // Real rows: single-exp pass (store e, then scale); cols >= TN forced to 0
            // so padded V rows contribute exactly 0. Padded query rows: all-zero
            // (their ctx/x stay finite and never reach real outputs).
#pragma unroll
            for (int rr = 0; rr < 2; ++rr) {
                int row = (lane << 1) + rr;
                float* sr = &sb[row * SS];
                if (row < TN) {
                    float mx = -1e30f;
                    for (int k = 0; k < TN; ++k) mx = fmaxf(mx, sr[k]);
                    float sum = 0.f;
                    for (int k = 0; k < TN; ++k) {
                        float e = expf(sr[k] - mx);
                        sr[k] = e;
                        sum += e;
                    }
                    float inv = 1.0f / sum;
                    for (int k = 0; k < TN; ++k) sr[k] *= inv;
                    for (int k = TN; k < TP; ++k) sr[k] = 0.0f;
                } else {
                    for (int k = 0; k < TP; ++k) sr[k] = 0.0f;
                }
            }
            __syncthreads();

            // ctx_h = P(64x64) @ V_h(64x4), K = 64 -> 16 k-steps
#pragma unroll
            for (int mt = 0; mt < 4; ++mt) {
                v8f acc = {};
#pragma unroll
                for (int kk = 0; kk < 16; ++kk) {
                    v2f a = frag_a4(&sb[(mt * 16) * SS + kk * 4], SS, lane);
                    v2f bf;
                    {
                        int col = lane & 15, kb = (lane >> 4) << 1;
                        if (col < DHD) {
                            const float* vp = &qb[(kk * 4) * QS + 2 * DM + hd * 4 + col];
                            bf.x = vp[kb * QS]; bf.y = vp[(kb + 1) * QS];
                        } else { bf.x = 0.f; bf.y = 0.f; }
                    }
                    acc = WMMA4(a, bf, acc);
                }
                int col = lane & 15, rb = (lane >> 4) << 3;
                if (col < DHD) {
#pragma unroll
                    for (int i = 0; i < 8; ++i)
                        hb[(mt * 16 + rb + i) * XS + hd * 4 + col] = acc[i];
                }
            }
            __syncthreads();
        }

        // ---------------- x += ctx @ Wout + bout (64x16 @ 16x16) ----------------
        {
            const float* wl = out_w + l * DM * DM;
            const float* bl = out_b + l * DM;
            int col = lane & 15, rb = (lane >> 4) << 3;
#pragma unroll
            for (int mt = 0; mt < 4; ++mt) {
                v8f acc;
                float bv = bl[col];
#pragma unroll
                for (int i = 0; i < 8; ++i)
                    acc[i] = xb[(mt * 16 + rb + i) * XS + col] + bv;
#pragma unroll
                for (int kk = 0; kk < 4; ++kk) {
                    v2f a = frag_a4(&hb[(mt * 16) * XS + kk * 4], XS, lane);
                    v2f bf = frag_b4(wl + (kk * 4) * DM, DM, lane);
                    acc = WMMA4(a, bf, acc);
                }
                frag_store(&xb[(mt * 16) * XS], XS, lane, acc);
            }
        }
        __syncthreads();

        // ---------------- LN2 ----------------
        layernorm_lds(xb, hb, ln2_s + l * DM, ln2_b + l * DM, lane);
        __syncthreads();

        // ---------------- ff1: gelu(h2 @ W1 + b1)  (64x16 @ 16x64) ----------------
        {
            const float* wl = ff1_w + l * DM * NFF;
            const float* bl = ff1_b + l * NFF;
            int col = lane & 15, rb = (lane >> 4) << 3;
#pragma unroll
            for (int mt = 0; mt < 4; ++mt)
#pragma unroll
                for (int nt = 0; nt < 4; ++nt) {
                    v8f acc;
                    float bv = bl[nt * 16 + col];
#pragma unroll
                    for (int i = 0; i < 8; ++i) acc[i] = bv;
#pragma unroll
                    for (int kk = 0; kk < 4; ++kk) {
                        v2f a = frag_a4(&hb[(mt * 16) * XS + kk * 4], XS, lane);
                        v2f bf = frag_b4(wl + (kk * 4) * NFF + nt * 16, NFF, lane);
                        acc = WMMA4(a, bf, acc);
                    }
#pragma unroll
                    for (int i = 0; i < 8; ++i)
                        sb[(mt * 16 + rb + i) * SS + nt * 16 + col] = gelu_exact(acc[i]);
                }
        }
        __syncthreads();

        // ---------------- x += a1 @ W2 + b2  (64x64 @ 64x16) ----------------
        {
            const float* wl = ff2_w + l * NFF * DM;
            const float* bl = ff2_b + l * DM;
            int col = lane & 15, rb = (lane >> 4) << 3;
#pragma unroll
            for (int mt = 0; mt < 4; ++mt) {
                v8f acc;
                float bv = bl[col];
#pragma unroll
                for (int i = 0; i < 8; ++i)
                    acc[i] = xb[(mt * 16 + rb + i) * XS + col] + bv;
#pragma unroll
                for (int kk = 0; kk < 16; ++kk) {
                    v2f a = frag_a4(&sb[(mt * 16) * SS + kk * 4], SS, lane);
                    v2f bf = frag_b4(wl + (kk * 4) * DM, DM, lane);
                    acc = WMMA4(a, bf, acc);
                }
                frag_store(&xb[(mt * 16) * XS], XS, lane, acc);
            }
        }
        __syncthreads();
    }

    // ---- stash cls row (token 0) ----
    if (lane < DM) cls_out[n * DM + lane] = xb[lane];
}

// ============ Kernel 2: head (16 -> 256) with coalesced writes over c ============
__global__ void __launch_bounds__(128)
head_kernel(const float* __restrict__ cls,   // (N,16)
            const float* __restrict__ w,     // (16,256)
            const float* __restrict__ bias,  // (256)
            float* __restrict__ out)         // (B,256,C)
{
    __shared__ float wl[DM * NHID];
    __shared__ float bl[NHID];
    const int tid = threadIdx.x;
    const int b   = blockIdx.x;
    const int c   = blockIdx.y * 128 + tid;

    for (int i = tid; i < DM * NHID; i += 128) wl[i] = w[i];
    for (int i = tid; i < NHID; i += 128) bl[i] = bias[i];
    __syncthreads();

    const int n = b * NC + c;
    float cr[DM];
#pragma unroll
    for (int d = 0; d < DM; ++d) cr[d] = cls[n * DM + d];

    for (int hid = 0; hid < NHID; ++hid) {
        float acc = bl[hid];
#pragma unroll
        for (int d = 0; d < DM; ++d) acc += cr[d] * wl[d * NHID + hid];
        out[(b * NHID + hid) * NC + c] = acc;   // consecutive lanes -> consecutive c
    }
}

extern "C" void kernel_launch(void* const* d_in, const int* in_sizes, int n_in,
                              void* d_out, int out_size, void* d_ws, size_t ws_size,
                              hipStream_t stream) {
    const float* couple    = (const float*)d_in[0];
    const float* feat_emb  = (const float*)d_in[1];
    const float* feat_bias = (const float*)d_in[2];
    const float* cls_tok   = (const float*)d_in[3];
    const float* track_emb = (const float*)d_in[4];
    const int*   role_ids  = (const int*)  d_in[5];
    const float* ln1_s = (const float*)d_in[6];
    const float* ln1_b = (const float*)d_in[7];
    const float* qkv_w = (const float*)d_in[8];
    const float* qkv_b = (const float*)d_in[9];
    const float* out_w = (const float*)d_in[10];
    const float* out_b = (const float*)d_in[11];
    const float* ln2_s = (const float*)d_in[12];
    const float* ln2_b = (const float*)d_in[13];
    const float* ff1_w = (const float*)d_in[14];
    const float* ff1_b = (const float*)d_in[15];
    const float* ff2_w = (const float*)d_in[16];
    const float* ff2_b = (const float*)d_in[17];
    const float* c2h_w = (const float*)d_in[18];
    const float* c2h_b = (const float*)d_in[19];

    float* cls_ws = (float*)d_ws;   // N*16 floats = 2 MB staging

    const int N = NB * NC;          // 32768 sequences, one wave32 each
    seq_encoder_kernel<<<N, 32, 0, stream>>>(
        couple, feat_emb, feat_bias, cls_tok, track_emb, role_ids,
        ln1_s, ln1_b, qkv_w, qkv_b, out_w, out_b,
        ln2_s, ln2_b, ff1_w, ff1_b, ff2_w, ff2_b, cls_ws);

    head_kernel<<<dim3(NB, NC / 128), 128, 0, stream>>>(cls_ws, c2h_w, c2h_b, (float*)d_out);
}